// MAE_CalcLoss_Raw_35751307772082
// MI455X (gfx1250) — compile-verified
//
#include <hip/hip_runtime.h>
#include <stdint.h>

#define B_  64
#define S_  2048
#define D_  512
#define NM_ 1536
#define NU_ 512
#define ROWS_M (B_ * NM_)            // 98304
#define ROWS_T (B_ * (NM_ + NU_))    // 131072
#define NBLK 2048
#define TPB  256
#define ITER (ROWS_T / NBLK)         // 64 rows per block
#define PIPE 4                       // async pipeline depth (3 rows in flight)

typedef __attribute__((ext_vector_type(2))) float v2f;
typedef __attribute__((ext_vector_type(8))) float v8f;

__global__ __launch_bounds__(TPB) void mae_partial_kernel(
    const float* __restrict__ outputs, const float* __restrict__ orig,
    const long long* __restrict__ mid, const long long* __restrict__ uid,
    float* __restrict__ part) {
  __shared__ __align__(16) float lp[PIPE][D_];   // pred rows, 4-stage pipeline
  __shared__ __align__(16) float lt[PIPE][D_];   // tgt rows,  4-stage pipeline
  __shared__ float red[TPB / 32];

  const int tid = threadIdx.x;
  const int blk = blockIdx.x;

  v2f accm = {0.0f, 0.0f};
  v2f accu = {0.0f, 0.0f};

  // Issue one gathered row (pred 2KB + tgt 2KB) into an LDS stage via the
  // CDNA5 async global->LDS engine (tracked by ASYNCcnt).
  auto issue = [&](int row, int stage) {
    int b; long long idx;
    if (row < ROWS_M) { b = row / NM_; idx = mid[row]; }
    else { int r = row - ROWS_M; b = r / NU_; idx = uid[r]; }
    const float* pr = outputs + ((size_t)b * S_ + (size_t)idx) * D_;
    const float* tg = orig    + ((size_t)b * S_ + (size_t)idx) * D_;
    if (tid < 128) {
      uint32_t l = (uint32_t)(uintptr_t)(&lp[stage][0]) + (uint32_t)tid * 16u;
      uint64_t g = (uint64_t)(uintptr_t)pr + (uint64_t)tid * 16u;
      asm volatile("global_load_async_to_lds_b128 %0, %1, off"
                   :: "v"(l), "v"(g) : "memory");
    } else {
      int t2 = tid - 128;
      uint32_t l = (uint32_t)(uintptr_t)(&lt[stage][0]) + (uint32_t)t2 * 16u;
      uint64_t g = (uint64_t)(uintptr_t)tg + (uint64_t)t2 * 16u;
      asm volatile("global_load_async_to_lds_b128 %0, %1, off"
                   :: "v"(l), "v"(g) : "memory");
    }
  };

  // Prime PIPE-1 stages.
  #pragma unroll
  for (int s = 0; s < PIPE - 1; ++s) issue(blk + s * NBLK, s);

  for (int k = 0; k < ITER; ++k) {
    const int row = blk + k * NBLK;

    if (k + PIPE - 1 < ITER)
      issue(blk + (k + PIPE - 1) * NBLK, (k + PIPE - 1) & (PIPE - 1));

    // Wait until row k has landed: allow exactly the newer rows to remain.
    const int rem = (ITER - 1 - k) < (PIPE - 1) ? (ITER - 1 - k) : (PIPE - 1);
    if      (rem >= 3) asm volatile("s_wait_asynccnt 0x3" ::: "memory");
    else if (rem == 2) asm volatile("s_wait_asynccnt 0x2" ::: "memory");
    else if (rem == 1) asm volatile("s_wait_asynccnt 0x1" ::: "memory");
    else               asm volatile("s_wait_asynccnt 0x0" ::: "memory");
    __syncthreads();                               // all waves' DMA visible

    const int st = k & (PIPE - 1);
    const v2f* vp = (const v2f*)&lp[st][0];
    const v2f* vt = (const v2f*)&lt[st][0];
    v2f p = vp[tid];
    v2f t = vt[tid];
    v2f d = p - t;
    if (row < ROWS_M) accm += d * d; else accu += d * d;

    __syncthreads();                               // reads done before reissue
  }

  // Fold in loss weights: mean over (B * n_idx * D), alpha on unmasked.
  const float sm = 1.0f / (float)((long long)B_ * NM_ * D_);
  const float su = 0.1f / (float)((long long)B_ * NU_ * D_);
  v2f tot = accm * sm + accu * su;

  // Wave reduction via the matrix pipe: A(16x4) = 64 lane partials, B = ones,
  // D rows hold partial sums -> 8 adds + one xor-16 combine.
  v8f c = {0.f, 0.f, 0.f, 0.f, 0.f, 0.f, 0.f, 0.f};
  v2f onesv = {1.0f, 1.0f};
  v8f dm = __builtin_amdgcn_wmma_f32_16x16x4_f32(
      false, tot, false, onesv, (short)0, c, false, false);
  float ls = dm[0] + dm[1] + dm[2] + dm[3] + dm[4] + dm[5] + dm[6] + dm[7];
  ls += __shfl_xor(ls, 16, 32);                   // lanes 0-15 + lanes 16-31

  const int lane = tid & 31, wid = tid >> 5;
  if (lane == 0) red[wid] = ls;
  __syncthreads();
  if (tid == 0) {
    float s = 0.0f;
    for (int i = 0; i < TPB / 32; ++i) s += red[i];
    part[blk] = s;
  }
}

__global__ __launch_bounds__(256) void mae_finalize_kernel(
    const float* __restrict__ part, float* __restrict__ out) {
  __shared__ float s[256];
  const int t = threadIdx.x;
  float a = 0.0f;
  for (int i = t; i < NBLK; i += 256) a += part[i];
  s[t] = a;
  __syncthreads();
  for (int off = 128; off > 0; off >>= 1) {
    if (t < off) s[t] += s[t + off];
    __syncthreads();
  }
  if (t == 0) out[0] = s[0];
}

extern "C" void kernel_launch(void* const* d_in, const int* in_sizes, int n_in,
                              void* d_out, int out_size, void* d_ws, size_t ws_size,
                              hipStream_t stream) {
  (void)in_sizes; (void)n_in; (void)out_size; (void)ws_size;
  const float*     outputs = (const float*)d_in[0];
  const float*     orig    = (const float*)d_in[1];
  const long long* mid     = (const long long*)d_in[2];
  const long long* uid     = (const long long*)d_in[3];
  float* part = (float*)d_ws;   // 2048 floats of scratch

  mae_partial_kernel<<<NBLK, TPB, 0, stream>>>(outputs, orig, mid, uid, part);
  mae_finalize_kernel<<<1, 256, 0, stream>>>(part, (float*)d_out);
}